// GNN_57277683859885
// MI455X (gfx1250) — compile-verified
//
#include <hip/hip_runtime.h>
#include <hip/hip_bf16.h>

typedef __attribute__((ext_vector_type(2))) float v2f;
typedef __attribute__((ext_vector_type(8))) float v8f;

#define N_NODES 50000
#define N_EDGES 800000
#define D 128
#define N_CLASSES 40
#define ROW_TILES 3125   // 50000 / 16, exact

// ---------------------------------------------------------------------------
// Degree / normalization kernels
// ---------------------------------------------------------------------------
__global__ __launch_bounds__(256) void deg_init(float* __restrict__ deg) {
    int i = blockIdx.x * blockDim.x + threadIdx.x;
    if (i < N_NODES) deg[i] = 1.0f;   // self-loop
}

__global__ __launch_bounds__(256) void deg_count(const int* __restrict__ dst,
                                                 float* __restrict__ deg) {
    int e = blockIdx.x * blockDim.x + threadIdx.x;
    if (e < N_EDGES) unsafeAtomicAdd(&deg[dst[e]], 1.0f);
}

__global__ __launch_bounds__(256) void deg_rsqrt(float* __restrict__ deg) {
    int i = blockIdx.x * blockDim.x + threadIdx.x;
    if (i < N_NODES) deg[i] = rsqrtf(deg[i]);
}

// ---------------------------------------------------------------------------
// H = A[N_NODES,128] @ W[128,128] via V_WMMA_F32_16X16X4_F32.
// 8 waves / block; each wave owns one 16-row tile and all 128 output cols.
// W staged to LDS K-pair-interleaved: Ws[(k/2)*256 + n*2 + (k&1)], so each
// B fragment (W[k][n], W[k+1][n]) is one contiguous 8B ds_load_b64 into an
// even-aligned VGPR pair -- no repacking moves before the WMMA.
// ---------------------------------------------------------------------------
__global__ __launch_bounds__(256) void gcn_gemm128(const float* __restrict__ A,
                                                   const float* __restrict__ W,
                                                   float* __restrict__ O) {
    __shared__ float Ws[D * D];   // 64 KB, interleaved layout
    const int tid = threadIdx.x;
    // stage + interleave: 64 k-pairs x 32 n-quads = 2048 units, 8 per thread
#pragma unroll
    for (int u = tid; u < 2048; u += 256) {
        const int kp = u >> 5;            // k-pair index 0..63  (k = 2*kp)
        const int n4 = (u & 31) * 4;      // n base 0,4,...,124
        const float4 r0 = *(const float4*)(W + (2 * kp) * D + n4);
        const float4 r1 = *(const float4*)(W + (2 * kp + 1) * D + n4);
        float4* dst = (float4*)(Ws + kp * 256 + n4 * 2);
        dst[0] = make_float4(r0.x, r1.x, r0.y, r1.y);
        dst[1] = make_float4(r0.z, r1.z, r0.w, r1.w);
    }
    __syncthreads();

    const int wave = tid >> 5;
    const int lane = tid & 31;
    const int tile = blockIdx.x * 8 + wave;     // wave-uniform
    if (tile >= ROW_TILES) return;              // uniform branch: EXEC stays full

    const int nlo   = lane & 15;
    const int khalf = lane >> 4;                // 0 or 1
    const int row   = tile * 16 + nlo;          // A-fragment row (M = lane%16)

    v8f acc[8] = {};                            // 8 N-tiles of 16x16 f32

    for (int k0 = 0; k0 < D; k0 += 4) {
        const int ka = k0 + 2 * khalf;          // even; lanes 0-15: K=k0,k0+1 ; 16-31: K=k0+2,k0+3
        v2f a;
        a.x = A[row * D + ka];
        a.y = A[row * D + ka + 1];
        const v2f* Brow = (const v2f*)(Ws + (ka >> 1) * 256);
#pragma unroll
        for (int t = 0; t < 8; ++t) {
            const v2f b = Brow[t * 16 + nlo];   // single ds_load_b64
            acc[t] = __builtin_amdgcn_wmma_f32_16x16x4_f32(
                false, a, false, b, (short)0, acc[t], false, false);
        }
    }

    // C/D layout: VGPR r -> M = r + 8*(lane>=16), N = lane%16
    const int rbase = tile * 16 + 8 * khalf;
#pragma unroll
    for (int t = 0; t < 8; ++t) {
        const int n = t * 16 + nlo;
#pragma unroll
        for (int r = 0; r < 8; ++r) O[(rbase + r) * D + n] = acc[t][r];
    }
}

// ---------------------------------------------------------------------------
// Zero the aggregation buffer (float4 granularity)
// ---------------------------------------------------------------------------
__global__ __launch_bounds__(256) void zero_buf(float4* __restrict__ p, int n4) {
    int i = blockIdx.x * blockDim.x + threadIdx.x;
    if (i < n4) p[i] = make_float4(0.f, 0.f, 0.f, 0.f);
}

// ---------------------------------------------------------------------------
// Edge scatter: agg[dst] += lin[src] * dinv[src]*dinv[dst].
// One wave per edge; lane handles 4 consecutive features (float4 gather,
// L2-resident on the 192MB L2, fp32 hardware atomics on scatter).
// ---------------------------------------------------------------------------
__global__ __launch_bounds__(256) void gcn_scatter(const float* __restrict__ lin,
                                                   const int* __restrict__ src,
                                                   const int* __restrict__ dst,
                                                   const float* __restrict__ dinv,
                                                   float* __restrict__ agg) {
    const int e    = (blockIdx.x * 256 + threadIdx.x) >> 5;
    const int lane = threadIdx.x & 31;
    if (e >= N_EDGES) return;
    const int s = src[e];
    const int d = dst[e];
    const float w = dinv[s] * dinv[d];
    const float4 v = ((const float4*)(lin + s * D))[lane];
    float* ap = agg + d * D + lane * 4;
    unsafeAtomicAdd(ap + 0, v.x * w);
    unsafeAtomicAdd(ap + 1, v.y * w);
    unsafeAtomicAdd(ap + 2, v.z * w);
    unsafeAtomicAdd(ap + 3, v.w * w);
}

// ---------------------------------------------------------------------------
// Epilogue: h = relu(agg + lin*dinv^2 (self loop) + bias)
// ---------------------------------------------------------------------------
__global__ __launch_bounds__(256) void gcn_epilogue(const float* __restrict__ agg,
                                                    const float* __restrict__ lin,
                                                    const float* __restrict__ dinv,
                                                    const float* __restrict__ bias,
                                                    float* __restrict__ hout) {
    int i = blockIdx.x * blockDim.x + threadIdx.x;   // over N_NODES*32 float4
    if (i >= N_NODES * (D / 4)) return;
    const int node = i >> 5;
    const int f4   = i & 31;
    float s = dinv[node];
    s = s * s;
    const float4 a = ((const float4*)agg)[i];
    const float4 l = ((const float4*)lin)[i];
    const float4 b = ((const float4*)bias)[f4];
    float4 r;
    r.x = fmaxf(a.x + l.x * s + b.x, 0.f);
    r.y = fmaxf(a.y + l.y * s + b.y, 0.f);
    r.z = fmaxf(a.z + l.z * s + b.z, 0.f);
    r.w = fmaxf(a.w + l.w * s + b.w, 0.f);
    ((float4*)hout)[i] = r;
}

// ---------------------------------------------------------------------------
// Final FC: logits = H[N,128] @ fcW[128,40] + fcb, via WMMA (3 col tiles of 16,
// cols 40..47 computed but never stored; B loads clamped in-bounds).
// ---------------------------------------------------------------------------
__global__ __launch_bounds__(256) void gcn_fc(const float* __restrict__ H,
                                              const float* __restrict__ W,
                                              const float* __restrict__ bias,
                                              float* __restrict__ O) {
    const int wave = threadIdx.x >> 5;
    const int lane = threadIdx.x & 31;
    const int tile = blockIdx.x * 8 + wave;
    if (tile >= ROW_TILES) return;

    const int nlo   = lane & 15;
    const int khalf = lane >> 4;
    const int row   = tile * 16 + nlo;

    v8f acc[3] = {};
    for (int k0 = 0; k0 < D; k0 += 4) {
        const int ka = k0 + 2 * khalf;
        v2f a;
        a.x = H[row * D + ka];
        a.y = H[row * D + ka + 1];
#pragma unroll
        for (int t = 0; t < 3; ++t) {
            const int n  = t * 16 + nlo;
            const int nc = n < N_CLASSES ? n : (N_CLASSES - 1);   // clamp load in-bounds
            v2f b;
            b.x = W[ka * N_CLASSES + nc];
            b.y = W[(ka + 1) * N_CLASSES + nc];
            acc[t] = __builtin_amdgcn_wmma_f32_16x16x4_f32(
                false, a, false, b, (short)0, acc[t], false, false);
        }
    }

    const int rbase = tile * 16 + 8 * khalf;
#pragma unroll
    for (int t = 0; t < 3; ++t) {
        const int n = t * 16 + nlo;
        if (n < N_CLASSES) {
            const float bb = bias[n];
#pragma unroll
            for (int r = 0; r < 8; ++r)
                O[(rbase + r) * N_CLASSES + n] = acc[t][r] + bb;
        }
    }
}

// ---------------------------------------------------------------------------
extern "C" void kernel_launch(void* const* d_in, const int* in_sizes, int n_in,
                              void* d_out, int out_size, void* d_ws, size_t ws_size,
                              hipStream_t stream) {
    const float* x   = (const float*)d_in[0];
    const int*   ei  = (const int*)d_in[1];
    const float* W1  = (const float*)d_in[2];
    const float* b1  = (const float*)d_in[3];
    const float* W2  = (const float*)d_in[4];
    const float* b2  = (const float*)d_in[5];
    const float* W3  = (const float*)d_in[6];
    const float* b3  = (const float*)d_in[7];
    const float* fcW = (const float*)d_in[8];
    const float* fcb = (const float*)d_in[9];

    float* out    = (float*)d_out;
    float* h_out  = out;                       // [N_NODES,128]
    float* logits = out + N_NODES * D;         // [N_NODES,40]

    float* ws   = (float*)d_ws;
    float* dinv = ws;                          // N_NODES (deg -> dinv in place)
    float* B0   = ws + 50176;                  // lin buffer  [N,128]
    float* B1   = B0 + N_NODES * D;            // agg/h ping  [N,128]

    const int* src = ei;
    const int* dst = ei + N_EDGES;

    const int TPB = 256;
    const int gN    = (N_NODES + TPB - 1) / TPB;
    const int gE    = (N_EDGES + TPB - 1) / TPB;
    const int gF4   = (N_NODES * (D / 4) + TPB - 1) / TPB;   // elementwise grids
    const int gTile = (ROW_TILES + 7) / 8;                   // 8 waves / block
    const int gScat = (N_EDGES + 7) / 8;                     // 1 wave / edge

    // normalization: deg -> dinv
    deg_init<<<gN, TPB, 0, stream>>>(dinv);
    deg_count<<<gE, TPB, 0, stream>>>(dst, dinv);
    deg_rsqrt<<<gN, TPB, 0, stream>>>(dinv);

    // layer 1: in = x
    gcn_gemm128<<<gTile, TPB, 0, stream>>>(x, W1, B0);
    zero_buf<<<gF4, TPB, 0, stream>>>((float4*)B1, N_NODES * (D / 4));
    gcn_scatter<<<gScat, TPB, 0, stream>>>(B0, src, dst, dinv, B1);
    gcn_epilogue<<<gF4, TPB, 0, stream>>>(B1, B0, dinv, b1, B1);

    // layer 2: in = B1 (gemm reads B1 before it is reused as agg)
    gcn_gemm128<<<gTile, TPB, 0, stream>>>(B1, W2, B0);
    zero_buf<<<gF4, TPB, 0, stream>>>((float4*)B1, N_NODES * (D / 4));
    gcn_scatter<<<gScat, TPB, 0, stream>>>(B0, src, dst, dinv, B1);
    gcn_epilogue<<<gF4, TPB, 0, stream>>>(B1, B0, dinv, b2, B1);

    // layer 3: in = B1, h result straight into d_out
    gcn_gemm128<<<gTile, TPB, 0, stream>>>(B1, W3, B0);
    zero_buf<<<gF4, TPB, 0, stream>>>((float4*)B1, N_NODES * (D / 4));
    gcn_scatter<<<gScat, TPB, 0, stream>>>(B0, src, dst, dinv, B1);
    gcn_epilogue<<<gF4, TPB, 0, stream>>>(B1, B0, dinv, b3, h_out);

    // final FC
    gcn_fc<<<gTile, TPB, 0, stream>>>(h_out, fcW, fcb, logits);
}